// TransMatch_17566416241101
// MI455X (gfx1250) — compile-verified
//
#include <hip/hip_runtime.h>

typedef float v2f __attribute__((ext_vector_type(2)));
typedef float v8f __attribute__((ext_vector_type(8)));

#define D     64
#define S     16
#define NREL  16
#define CSTR  17   // padded stride for count matrix (LDS bank spread)

__global__ __launch_bounds__(256) void transmatch_kernel(
    const int*   __restrict__ relations,      // (BS,)
    const int*   __restrict__ entity_pairs,   // (BS,2)
    const int*   __restrict__ train_edges,    // (BS,)
    const int*   __restrict__ entity2edges,   // (N_ENT,S)
    const int*   __restrict__ edge2entities,  // (N_EDGE,2)
    const int*   __restrict__ edge2relation,  // (N_EDGE,)
    const float* __restrict__ relation_emb,   // (NREL,D)
    const float* __restrict__ entity_emb,     // (N_ENT,D)
    const float* __restrict__ scorer_w,       // (D,)
    const float* __restrict__ scorer_b,       // (1,)
    float*       __restrict__ out)            // (BS,)
{
    __shared__ float s_relemb[NREL * D];   // 4 KB relation table
    __shared__ float s_w[D];
    __shared__ int   s_e1[32];             // hop-1 edges
    __shared__ float s_m1[32];             // hop-1 mask
    __shared__ int   s_r1[32];             // hop-1 relation ids
    __shared__ int   s_ent[64];            // hop-1 tail entities (32 edges x 2)
    __shared__ float s_cnt[64 * CSTR];     // per-position relation counts (A matrix)
    __shared__ float s_cntsum[64];         // masked neighbor counts (>=1)
    __shared__ float s_aggB[64 * D];       // WMMA result: counts @ relemb
    __shared__ float s_ev1[32 * D];        // updated hop-1 edge vectors
    __shared__ float s_part[2 * D];
    __shared__ float s_red[D];

    const int b   = blockIdx.x;
    const int tid = threadIdx.x;
    const int te  = train_edges[b];

    // ---- stage 0: cache relation table + scorer weights in LDS ----
    for (int i = tid; i < NREL * D; i += 256) s_relemb[i] = relation_emb[i];
    if (tid < D) s_w[tid] = scorer_w[tid];

    // ---- stage 1: hop-1 edges, mask, relation ids ----
    if (tid < 32) {
        int ent0 = entity_pairs[b * 2 + (tid >> 4)];
        int e1   = entity2edges[(size_t)ent0 * S + (tid & 15)];
        s_e1[tid] = e1;
        s_m1[tid] = (e1 != te) ? 1.0f : 0.0f;
        s_r1[tid] = edge2relation[e1];
    }
    __syncthreads();

    // ---- stage 2: hop-1 tail entities ----
    if (tid < 64) {
        s_ent[tid] = edge2entities[(size_t)s_e1[tid >> 1] * 2 + (tid & 1)];
    }
    __syncthreads();

    // ---- stage 3: hop-2 masked relation histogram (one position per thread) ----
    if (tid < 64) {
        const int ent = s_ent[tid];
        float* crow = &s_cnt[tid * CSTR];
        #pragma unroll
        for (int r = 0; r < NREL; ++r) crow[r] = 0.0f;
        float cnt = 0.0f;
        #pragma unroll
        for (int s = 0; s < S; ++s) {
            int e2 = entity2edges[(size_t)ent * S + s];
            if (e2 != te) {
                int rid = edge2relation[e2];
                crow[rid] += 1.0f;
                cnt += 1.0f;
            }
        }
        s_cntsum[tid] = (cnt > 0.0f) ? cnt : 1.0f;
    }
    __syncthreads();

    // ---- stage 4: aggB = counts(64x16) @ relemb(16x64) via V_WMMA_F32_16X16X4_F32 ----
    // 8 waves: wave w -> row block (w>>1)*16, column half (w&1)*32 (two 16-col tiles).
    {
        const int wave  = tid >> 5;
        const int lane  = tid & 31;
        const int rb    = wave >> 1;        // position-row block (0..3)
        const int cb    = wave & 1;         // 32-dim column half
        const int l16   = lane & 15;
        const int khalf = lane >> 4;        // ISA A-layout: lanes16-31 hold K+2
        #pragma unroll
        for (int ct = 0; ct < 2; ++ct) {
            const int ncol = cb * 32 + ct * 16 + l16;
            v8f acc = {};
            #pragma unroll
            for (int kk = 0; kk < 4; ++kk) {            // K = 16 in steps of 4
                const int k0 = kk * 4 + khalf * 2;
                v2f a, bm;
                a.x  = s_cnt[(rb * 16 + l16) * CSTR + k0];
                a.y  = s_cnt[(rb * 16 + l16) * CSTR + k0 + 1];
                bm.x = s_relemb[(k0    ) * D + ncol];
                bm.y = s_relemb[(k0 + 1) * D + ncol];
                acc = __builtin_amdgcn_wmma_f32_16x16x4_f32(
                        false, a, false, bm, (short)0, acc, false, false);
            }
            #pragma unroll
            for (int v = 0; v < 8; ++v) {
                const int row = rb * 16 + v + ((lane >= 16) ? 8 : 0);
                s_aggB[row * D + ncol] = acc[v];
            }
        }
    }
    __syncthreads();

    // ---- stage 5: updated hop-1 edge vectors ev1' ----
    // ev1'[j] = relemb[r1[j]] + 0.5*(ee0 + agg0 + ee1 + agg1)
    {
        const int d  = tid & 63;
        const int jg = tid >> 6;            // 4 rows per iteration
        for (int it = 0; it < 8; ++it) {
            const int j  = it * 4 + jg;
            const int p0 = j * 2, p1 = j * 2 + 1;
            float ee0 = entity_emb[(size_t)s_ent[p0] * D + d];
            float ee1 = entity_emb[(size_t)s_ent[p1] * D + d];
            float a0  = s_aggB[p0 * D + d] / s_cntsum[p0];
            float a1  = s_aggB[p1 * D + d] / s_cntsum[p1];
            s_ev1[j * D + d] =
                s_relemb[s_r1[j] * D + d] + 0.5f * (ee0 + a0 + ee1 + a1);
        }
    }
    __syncthreads();

    // ---- stage 6: hop-0 aggregation (uses original relemb[r1] for i=0, ev1' for i=1) ----
    if (tid < 128) {
        const int k = tid >> 6;
        const int d = tid & 63;
        float sA = 0.0f, sC = 0.0f, cnt1 = 0.0f;
        #pragma unroll
        for (int s = 0; s < S; ++s) {
            const int idx = k * S + s;
            const float m = s_m1[idx];
            cnt1 += m;
            sA += m * s_relemb[s_r1[idx] * D + d];
            sC += m * s_ev1[idx * D + d];
        }
        if (cnt1 == 0.0f) cnt1 = 1.0f;
        float ee = entity_emb[(size_t)entity_pairs[b * 2 + k] * D + d];
        s_part[k * D + d] = ee + 0.5f * (sA + sC) / cnt1;
    }
    __syncthreads();

    // ---- stage 7: final vector + score ----
    if (tid < 64) {
        const int rel = relations[b];
        float o = s_relemb[rel * D + tid] + s_part[tid] + s_part[D + tid];
        s_red[tid] = o * s_w[tid];
    }
    __syncthreads();
    if (tid == 0) {
        float acc = scorer_b[0];
        #pragma unroll
        for (int d = 0; d < D; ++d) acc += s_red[d];
        out[b] = acc;
    }
}

extern "C" void kernel_launch(void* const* d_in, const int* in_sizes, int n_in,
                              void* d_out, int out_size, void* d_ws, size_t ws_size,
                              hipStream_t stream) {
    const int*   relations     = (const int*)d_in[0];
    const int*   entity_pairs  = (const int*)d_in[1];
    const int*   train_edges   = (const int*)d_in[2];
    const int*   entity2edges  = (const int*)d_in[3];
    const int*   edge2entities = (const int*)d_in[4];
    const int*   edge2relation = (const int*)d_in[5];
    const float* relation_emb  = (const float*)d_in[6];
    const float* entity_emb    = (const float*)d_in[7];
    const float* scorer_w      = (const float*)d_in[8];
    const float* scorer_b      = (const float*)d_in[9];
    float*       out           = (float*)d_out;

    const int bs = in_sizes[0];   // 1024
    transmatch_kernel<<<bs, 256, 0, stream>>>(
        relations, entity_pairs, train_edges, entity2edges, edge2entities,
        edge2relation, relation_emb, entity_emb, scorer_w, scorer_b, out);
}